// VectorQuantizerEMA_70403103916641
// MI455X (gfx1250) — compile-verified
//
#include <hip/hip_runtime.h>
#include <hip/hip_bf16.h>
#include <math.h>

#define D_DIM  256
#define K_DIM  4096
#define N_ROWS 32768
#define DECAY  0.99f
#define EPS_C  1e-5f

typedef __attribute__((ext_vector_type(2))) float v2f;
typedef __attribute__((ext_vector_type(8))) float v8f;

// ---- workspace layout (float offsets) ----
#define WS_IDX     0                         // 32768 ints (aliased)
#define WS_COUNTS  32768                     // 4096 f
#define WS_ENORM   36864                     // 4096 f
#define WS_CNORM   40960                     // 4096 f
#define WS_LOSS    45056                     // 64 f (1 used)
#define WS_ET      45120                     // K*D f, [K][D]
#define WS_DWT     (45120 + K_DIM*D_DIM)     // K*D f, [K][D]

// ---- output layout (float offsets) ----
#define OUT_QUANT  0
#define OUT_LOSS   8388608
#define OUT_PERP   8388609
#define OUT_IDX    8388610
#define OUT_NEWE   8421378

#define A_STRIDE 260   // 256 + 4 pad -> bank-conflict-free A fragment reads
#define NKT      (K_DIM / 16)

// ---------------------------------------------------------------------------
// zero scratch accumulators (graph-capture safe; harness poisons ws once)
// ---------------------------------------------------------------------------
__global__ void __launch_bounds__(256) zero_ws_kernel(float* __restrict__ counts,
                                                      float* __restrict__ loss_acc,
                                                      float* __restrict__ dwT) {
  int i = blockIdx.x * 256 + threadIdx.x;          // grid covers D*K
  dwT[i] = 0.f;
  if (i < K_DIM) counts[i] = 0.f;
  if (i < 64)    loss_acc[i] = 0.f;
}

// ---------------------------------------------------------------------------
// enorm[k] = sum_d E[d][k]^2   (coalesced across k)
// ---------------------------------------------------------------------------
__global__ void __launch_bounds__(256) enorm_kernel(const float* __restrict__ E,
                                                    float* __restrict__ enorm) {
  int k = blockIdx.x * 256 + threadIdx.x;
  float s = 0.f;
  for (int d = 0; d < D_DIM; ++d) {
    float e = E[(size_t)d * K_DIM + k];
    s += e * e;
  }
  enorm[k] = s;
}

// ---------------------------------------------------------------------------
// ET[k][d] = E[d][k]   (tiled transpose so the gather reads contiguous rows)
// ---------------------------------------------------------------------------
__global__ void transpose_kernel(const float* __restrict__ E, float* __restrict__ ET) {
  __shared__ float tile[32][33];
  int k0 = blockIdx.x * 32, d0 = blockIdx.y * 32;
  for (int i = threadIdx.y; i < 32; i += 8)
    tile[i][threadIdx.x] = E[(size_t)(d0 + i) * K_DIM + k0 + threadIdx.x];
  __syncthreads();
  for (int i = threadIdx.y; i < 32; i += 8)
    ET[(size_t)(k0 + i) * D_DIM + d0 + threadIdx.x] = tile[threadIdx.x][i];
}

// ---------------------------------------------------------------------------
// Core kernel: fp32 WMMA score tiles + running argmin + dw scatter-add.
// Block = 256 threads = 8 waves; 128 rows per block staged in LDS once.
// B tiles (16 codes x 256 dims) are register-double-buffered: next tile's
// global loads are issued before the 64-WMMA loop (which hides the latency),
// stored to the alternate LDS buffer after compute, single barrier per tile.
// Score = ||e||^2 - 2*x.e  (||x||^2 is row-constant, irrelevant for argmin).
// ---------------------------------------------------------------------------
__global__ void __launch_bounds__(256) vq_argmin_wmma(
    const float* __restrict__ x, const float* __restrict__ E,
    const float* __restrict__ enorm, int* __restrict__ idx_out,
    float* __restrict__ counts, float* __restrict__ dwT)
{
  extern __shared__ float smem[];
  float* Als  = smem;                          // 128 * A_STRIDE
  float* Bls0 = Als + 128 * A_STRIDE;          // 256 * 16 (d-major)
  float* Bls1 = Bls0 + 256 * 16;               // 256 * 16
  float* Ens  = Bls1 + 256 * 16;               // 4096 (all of enorm)
  int*   widx = (int*)(Ens + K_DIM);           // 128

  const int tid  = threadIdx.x;
  const int wave = tid >> 5;
  const int lane = tid & 31;
  const int m    = lane & 15;                  // A row / B col / C col index
  const int h    = lane >> 4;                  // K-pair half select
  const int rowBase = blockIdx.x * 128;

  // stage A: 128 rows x 256 dims, float4 coalesced
  for (int i = tid; i < 128 * 64; i += 256) {
    int r = i >> 6, c4 = i & 63;
    const float4 v = reinterpret_cast<const float4*>(x + (size_t)(rowBase + r) * D_DIM)[c4];
    float* dst = Als + r * A_STRIDE + c4 * 4;
    dst[0] = v.x; dst[1] = v.y; dst[2] = v.z; dst[3] = v.w;
  }
  // stage full enorm (16 KB) -> kills the per-tile global load stall
  for (int i = tid; i < K_DIM / 4; i += 256) {
    const float4 v = reinterpret_cast<const float4*>(enorm)[i];
    float* dst = Ens + i * 4;
    dst[0] = v.x; dst[1] = v.y; dst[2] = v.z; dst[3] = v.w;
  }

  // preload B tile 0 into LDS buffer 0 (1024 float4 cooperative)
  float4 breg[4];
#pragma unroll
  for (int c = 0; c < 4; ++c) {
    int i = c * 256 + tid;
    int d = i >> 2, n4 = i & 3;
    breg[c] = reinterpret_cast<const float4*>(E + (size_t)d * K_DIM)[n4];
  }
#pragma unroll
  for (int c = 0; c < 4; ++c) {
    int i = c * 256 + tid;
    int d = i >> 2, n4 = i & 3;
    float* dst = Bls0 + d * 16 + n4 * 4;
    dst[0] = breg[c].x; dst[1] = breg[c].y; dst[2] = breg[c].z; dst[3] = breg[c].w;
  }
  __syncthreads();

  float minv[8];
  int   mini[8];
#pragma unroll
  for (int v = 0; v < 8; ++v) { minv[v] = 3.4e38f; mini[v] = 0; }

  // A fragment: lane holds A[M=m][d0+2h], A[M=m][d0+2h+1]
  const float* arow = Als + (wave * 16 + m) * A_STRIDE + 2 * h;

  for (int kt = 0; kt < NKT; ++kt) {
    float* cur = (kt & 1) ? Bls1 : Bls0;
    float* nxt = (kt & 1) ? Bls0 : Bls1;
    const bool more = (kt + 1) < NKT;

    // issue next tile's global loads now; the WMMA loop below hides them
    if (more) {
#pragma unroll
      for (int c = 0; c < 4; ++c) {
        int i = c * 256 + tid;
        int d = i >> 2, n4 = i & 3;
        breg[c] = reinterpret_cast<const float4*>(E + (size_t)d * K_DIM + (kt + 1) * 16)[n4];
        if (kt + 2 < NKT)  // warm WGP$ for tile kt+2 -> global_prefetch_b8
          __builtin_prefetch(E + (size_t)d * K_DIM + (kt + 2) * 16 + n4 * 4, 0, 0);
      }
    }

    // B fragment: lane holds B[d0+2h][N=m], B[d0+2h+1][N=m]
    const float* bcol = cur + (2 * h) * 16 + m;
    v8f acc = {};
#pragma unroll 8
    for (int d0 = 0; d0 < D_DIM; d0 += 4) {
      v2f a; a.x = arow[d0];        a.y = arow[d0 + 1];
      v2f b; b.x = bcol[d0 * 16];   b.y = bcol[d0 * 16 + 16];
      acc = __builtin_amdgcn_wmma_f32_16x16x4_f32(
          /*neg_a=*/false, a, /*neg_b=*/false, b,
          /*c_mod=*/(short)0, acc, /*reuse_a=*/false, /*reuse_b=*/false);
    }

    const int   code = kt * 16 + m;             // this lane's column code
    const float en   = Ens[code];
#pragma unroll
    for (int v = 0; v < 8; ++v) {               // C: lane holds rows v+8h, col m
      float s = en - 2.0f * acc[v];
      if (s < minv[v]) { minv[v] = s; mini[v] = code; }
    }

    // commit next tile to the alternate LDS buffer (prev readers done at the
    // barrier that ended iteration kt-1; cur readers are unaffected)
    if (more) {
#pragma unroll
      for (int c = 0; c < 4; ++c) {
        int i = c * 256 + tid;
        int d = i >> 2, n4 = i & 3;
        float* dst = nxt + d * 16 + n4 * 4;
        dst[0] = breg[c].x; dst[1] = breg[c].y; dst[2] = breg[c].z; dst[3] = breg[c].w;
      }
    }
    __syncthreads();                            // single barrier per tile
  }

  // reduce argmin across the 16 columns (lanes within each half, wave32 xor)
#pragma unroll
  for (int v = 0; v < 8; ++v) {
    float bv = minv[v]; int bi = mini[v];
#pragma unroll
    for (int off = 1; off < 16; off <<= 1) {
      float ov = __shfl_xor(bv, off, 32);
      int   oi = __shfl_xor(bi, off, 32);
      if (ov < bv || (ov == bv && oi < bi)) { bv = ov; bi = oi; }
    }
    minv[v] = bv; mini[v] = bi;
  }
  // lane 0 has rows 0..7, lane 16 has rows 8..15 of this wave's 16-row group
  if (m == 0) {
#pragma unroll
    for (int v = 0; v < 8; ++v) {
      int r = v + 8 * h;
      int g = rowBase + wave * 16 + r;
      idx_out[g]          = mini[v];
      widx[wave * 16 + r] = mini[v];
      atomicAdd(&counts[mini[v]], 1.0f);
    }
  }
  __syncthreads();

  // dw scatter: dwT[k][d] += x[row][d]  (rows still hot in LDS; contiguous d)
  for (int r = 0; r < 16; ++r) {
    int kc = widx[wave * 16 + r];
    const float* src = Als + (wave * 16 + r) * A_STRIDE;
    for (int d = lane; d < D_DIM; d += 32)
      atomicAdd(&dwT[(size_t)kc * D_DIM + d], src[d]);
  }
}

// ---------------------------------------------------------------------------
// gather quantized rows + squared-error partial sums + idx-as-float output
// one wave per row
// ---------------------------------------------------------------------------
__global__ void __launch_bounds__(256) gather_loss_kernel(
    const float* __restrict__ x, const float* __restrict__ ET,
    const int* __restrict__ idx, float* __restrict__ quant,
    float* __restrict__ idxf, float* __restrict__ loss_acc)
{
  int row  = blockIdx.x * 8 + (threadIdx.x >> 5);
  int lane = threadIdx.x & 31;
  int k = idx[row];
  const float* e  = ET + (size_t)k * D_DIM;
  const float* xr = x  + (size_t)row * D_DIM;
  float*       q  = quant + (size_t)row * D_DIM;
  float s = 0.f;
  for (int d = lane; d < D_DIM; d += 32) {
    float qv = e[d];
    q[d] = qv;
    float df = qv - xr[d];
    s += df * df;
  }
#pragma unroll
  for (int off = 16; off; off >>= 1) s += __shfl_xor(s, off, 32);
  if (lane == 0) {
    atomicAdd(loss_acc, s);
    idxf[row] = (float)k;
  }
}

// ---------------------------------------------------------------------------
// single-block cluster statistics: EMA counts, n-sum, cluster_norm, perplexity,
// vq_loss finalize
// ---------------------------------------------------------------------------
__global__ void __launch_bounds__(256) stats_kernel(
    const float* __restrict__ counts, const float* __restrict__ ema_cluster,
    const float* __restrict__ counter_p, const float* __restrict__ loss_acc,
    float* __restrict__ cnorm, float* __restrict__ out_loss,
    float* __restrict__ out_perp)
{
  __shared__ float s_n[256], s_e[256];
  int tid = threadIdx.x;
  float bias = 1.0f - powf(DECAY, counter_p[0] + 1.0f);
  float cav[16];
  float sn = 0.f, se = 0.f;
#pragma unroll
  for (int j = 0; j < 16; ++j) {
    int k = tid + j * 256;
    float c  = counts[k];
    float ch = ema_cluster[k] * DECAY + c * (1.0f - DECAY);
    float a  = ch / bias;
    cav[j] = a; sn += a;
    float p = c / (float)N_ROWS;
    se += p * logf(p + 1e-10f);
  }
  s_n[tid] = sn; s_e[tid] = se;
  __syncthreads();
  for (int s = 128; s > 0; s >>= 1) {
    if (tid < s) { s_n[tid] += s_n[tid + s]; s_e[tid] += s_e[tid + s]; }
    __syncthreads();
  }
  float n = s_n[0];
#pragma unroll
  for (int j = 0; j < 16; ++j) {
    int k = tid + j * 256;
    cnorm[k] = (cav[j] + EPS_C) / (n + K_DIM * EPS_C) * n;
  }
  if (tid == 0) {
    out_perp[0] = expf(-s_e[0]);
    out_loss[0] = 1.25f * loss_acc[0] / (float)((size_t)N_ROWS * D_DIM);  // q + COMMIT*e
  }
}

// ---------------------------------------------------------------------------
// new_embeddings[d][k] = ((ema_dw*DECAY + dw*(1-DECAY))/bias) / cluster_norm[k]
// ---------------------------------------------------------------------------
__global__ void __launch_bounds__(256) newemb_kernel(
    const float* __restrict__ ema_dw, const float* __restrict__ dwT,
    const float* __restrict__ cnorm, const float* __restrict__ counter_p,
    float* __restrict__ out)
{
  int i = blockIdx.x * 256 + threadIdx.x;   // over [D][K]
  float bias = 1.0f - powf(DECAY, counter_p[0] + 1.0f);
  int d = i >> 12;          // i / K
  int k = i & (K_DIM - 1);  // i % K
  float dwv = dwT[(size_t)k * D_DIM + d] * (1.0f - DECAY);
  float dwh = ema_dw[i] * DECAY + dwv;
  out[i] = dwh / bias / cnorm[k];
}

// ---------------------------------------------------------------------------
extern "C" void kernel_launch(void* const* d_in, const int* in_sizes, int n_in,
                              void* d_out, int out_size, void* d_ws, size_t ws_size,
                              hipStream_t stream) {
  (void)in_sizes; (void)n_in; (void)out_size; (void)ws_size;

  const float* x           = (const float*)d_in[0];
  const float* E           = (const float*)d_in[1];
  const float* counter     = (const float*)d_in[2];
  const float* ema_cluster = (const float*)d_in[3];
  const float* ema_dw      = (const float*)d_in[4];

  float* ws      = (float*)d_ws;
  int*   idx_i   = (int*)d_ws;              // WS_IDX == 0
  float* counts  = ws + WS_COUNTS;
  float* enorm   = ws + WS_ENORM;
  float* cnorm   = ws + WS_CNORM;
  float* lossacc = ws + WS_LOSS;
  float* ET      = ws + WS_ET;
  float* dwT     = ws + WS_DWT;

  float* out   = (float*)d_out;
  float* quant = out + OUT_QUANT;
  float* loss  = out + OUT_LOSS;
  float* perp  = out + OUT_PERP;
  float* idxf  = out + OUT_IDX;
  float* newe  = out + OUT_NEWE;

  zero_ws_kernel<<<(D_DIM * K_DIM) / 256, 256, 0, stream>>>(counts, lossacc, dwT);
  enorm_kernel<<<K_DIM / 256, 256, 0, stream>>>(E, enorm);
  transpose_kernel<<<dim3(K_DIM / 32, D_DIM / 32), dim3(32, 8), 0, stream>>>(E, ET);

  size_t smem_bytes =
      (size_t)(128 * A_STRIDE + 2 * 256 * 16 + K_DIM) * sizeof(float) + 128 * sizeof(int);
  vq_argmin_wmma<<<N_ROWS / 128, 256, smem_bytes, stream>>>(x, E, enorm, idx_i, counts, dwT);

  gather_loss_kernel<<<N_ROWS / 8, 256, 0, stream>>>(x, ET, idx_i, quant, idxf, lossacc);
  stats_kernel<<<1, 256, 0, stream>>>(counts, ema_cluster, counter, lossacc, cnorm, loss, perp);
  newemb_kernel<<<(D_DIM * K_DIM) / 256, 256, 0, stream>>>(ema_dw, dwT, cnorm, counter, newe);
}